// PointLayerNorm_85023172591851
// MI455X (gfx1250) — compile-verified
//
#include <hip/hip_runtime.h>

#define NROWS  131072
#define NDIM   512
#define NBATCH 16
#define LN_EPS 1e-5f

typedef __attribute__((ext_vector_type(2))) float v2f;
typedef __attribute__((ext_vector_type(4))) float v4f;
typedef __attribute__((ext_vector_type(8))) float v8f;

#if defined(__gfx1250__) && __has_builtin(__builtin_amdgcn_wmma_f32_16x16x4_f32)
#define USE_WMMA_F32 1
#else
#define USE_WMMA_F32 0
#endif

// ---------------------------------------------------------------------------
// Kernel 0: re-zero the 48 accumulator floats in workspace (graph-replay safe).
// ---------------------------------------------------------------------------
__global__ __launch_bounds__(64) void pln_zero(float* __restrict__ ws) {
    const unsigned t = threadIdx.x;
    if (t < 3u * NBATCH) ws[t] = 0.0f;
}

// ---------------------------------------------------------------------------
// Kernel 1: per-segment S1 = sum(x), S2 = sum(x^2), cnt. One wave = one
// 16-row tile. Lane m and lane m+16 stream the two 4-float-interleaved halves
// of row m with b128 loads. Flush uses WMMA f32 16x16x4 with 0/1 selector B
// matrices to produce exact per-row sums, then scatters by bidx into LDS.
// ---------------------------------------------------------------------------
__global__ __launch_bounds__(256) void pln_stats(const float* __restrict__ x,
                                                 const int*   __restrict__ bidx,
                                                 float* __restrict__ gs1,
                                                 float* __restrict__ gs2,
                                                 float* __restrict__ gcnt) {
    __shared__ float ls1[NBATCH], ls2[NBATCH], lcnt[NBATCH];
    const unsigned tid = threadIdx.x;
    if (tid < NBATCH) { ls1[tid] = 0.0f; ls2[tid] = 0.0f; lcnt[tid] = 0.0f; }
    __syncthreads();

    const unsigned wave = tid >> 5;          // 8 waves/block
    const unsigned lane = tid & 31u;
    const unsigned m    = lane & 15u;        // row within tile
    const unsigned half = lane >> 4;         // which 4-float interleave half
    const unsigned tile = blockIdx.x * 8u + wave;   // 8192 tiles total
    const unsigned row0 = tile * 16u;

    const float* rp = x + (size_t)(row0 + m) * NDIM + half * 4u;

    v4f a1 = {0.0f, 0.0f, 0.0f, 0.0f};
    v4f a2 = {0.0f, 0.0f, 0.0f, 0.0f};
#pragma unroll 4
    for (int c = 0; c < NDIM; c += 8) {
        const v4f v = *reinterpret_cast<const v4f*>(rp + c);
        if (c + 64 < NDIM) __builtin_prefetch(rp + c + 64, 0, 1);  // global_prefetch_b8
        a1.x += v.x; a1.y += v.y; a1.z += v.z; a1.w += v.w;
        a2.x = fmaf(v.x, v.x, a2.x);
        a2.y = fmaf(v.y, v.y, a2.y);
        a2.z = fmaf(v.z, v.z, a2.z);
        a2.w = fmaf(v.w, v.w, a2.w);
    }
    const float p1 = (a1.x + a1.y) + (a1.z + a1.w);   // half-row sum(x)
    const float p2 = (a2.x + a2.y) + (a2.z + a2.w);   // half-row sum(x^2)

#if USE_WMMA_F32
    // A-fragment (16x4 f32): A[m][0]=p1@lane m, A[m][2]=p1@lane m+16,
    //                        A[m][1]=p2@lane m, A[m][3]=p2@lane m+16.
    v2f A;  A.x = p1;   A.y = p2;
    v2f B1; B1.x = 1.0f; B1.y = 0.0f;   // select K={0,2}: full-row sum(x)
    v2f B2; B2.x = 0.0f; B2.y = 1.0f;   // select K={1,3}: full-row sum(x^2)
    v8f cz = {};
    v8f d1 = __builtin_amdgcn_wmma_f32_16x16x4_f32(false, A, false, B1,
                                                   (short)0, cz, false, false);
    v8f d2 = __builtin_amdgcn_wmma_f32_16x16x4_f32(false, A, false, B2,
                                                   (short)0, cz, false, false);
    // D layout: VGPR r, lanes 0-15 -> row r; lanes 16-31 -> row r+8.
    if (m == 0u) {
        const unsigned rbase = row0 + half * 8u;
#pragma unroll
        for (int r = 0; r < 8; ++r) {
            const int b = bidx[rbase + r];
            atomicAdd(&ls1[b], d1[r]);
            atomicAdd(&ls2[b], d2[r]);
            atomicAdd(&lcnt[b], 1.0f);
        }
    }
#else
    const float r1 = p1 + __shfl_down(p1, 16, 32);
    const float r2 = p2 + __shfl_down(p2, 16, 32);
    if (half == 0u) {
        const int b = bidx[row0 + m];
        atomicAdd(&ls1[b], r1);
        atomicAdd(&ls2[b], r2);
        atomicAdd(&lcnt[b], 1.0f);
    }
#endif
    __syncthreads();
    if (tid < NBATCH) {
        atomicAdd(&gs1[tid], ls1[tid]);
        atomicAdd(&gs2[tid], ls2[tid]);
        atomicAdd(&gcnt[tid], lcnt[tid]);
    }
}

// ---------------------------------------------------------------------------
// Kernel 2: mean / inv_std per segment (16 values).
// ---------------------------------------------------------------------------
__global__ void pln_finalize(float* __restrict__ ws) {
    const unsigned b = threadIdx.x;
    if (b < NBATCH) {
        const float cnt   = fmaxf(ws[32 + b], 1.0f);
        const float denom = cnt * (float)NDIM;
        const float mu    = ws[b] / denom;
        const float e2    = ws[16 + b] / denom;
        float       var   = fmaxf(e2 - mu * mu, 0.0f);
        ws[48 + b] = mu;
        ws[64 + b] = 1.0f / sqrtf(var + LN_EPS);
    }
}

// ---------------------------------------------------------------------------
// Kernel 3: out = (x - mean[b]) * inv[b] * w + bias. One wave per row.
// Blocks iterate in REVERSE row order so the tail of x left in the 192MB L2
// by pass 1 is hit first. x loads are nontemporal (read-once), out stores
// nontemporal (pure stream) to keep L2 for the resident x tail.
// ---------------------------------------------------------------------------
__global__ __launch_bounds__(256) void pln_norm(const float* __restrict__ x,
                                                const int*   __restrict__ bidx,
                                                const float* __restrict__ wgt,
                                                const float* __restrict__ bias,
                                                const float* __restrict__ mean,
                                                const float* __restrict__ inv,
                                                float* __restrict__ out) {
    __shared__ __align__(16) float wsh[NDIM];
    __shared__ __align__(16) float bsh[NDIM];
    const unsigned tid = threadIdx.x;
    wsh[tid]        = wgt[tid];
    wsh[tid + 256u] = wgt[tid + 256u];
    bsh[tid]        = bias[tid];
    bsh[tid + 256u] = bias[tid + 256u];
    __syncthreads();

    const unsigned blk  = gridDim.x - 1u - blockIdx.x;  // reverse traversal
    const unsigned wave = tid >> 5;
    const unsigned lane = tid & 31u;
    const unsigned row  = blk * 8u + wave;

    const int   b  = bidx[row];
    const float mu = mean[b];
    const float is = inv[b];

    const float* xr  = x   + (size_t)row * NDIM;
    float*       orp = out + (size_t)row * NDIM;
#pragma unroll
    for (int j = 0; j < 4; ++j) {
        const int c = (j * 32 + (int)lane) * 4;
        const v4f v = __builtin_nontemporal_load(reinterpret_cast<const v4f*>(xr + c));
        const v4f w = *reinterpret_cast<const v4f*>(&wsh[c]);
        const v4f g = *reinterpret_cast<const v4f*>(&bsh[c]);
        v4f o;
        o.x = fmaf((v.x - mu) * is, w.x, g.x);
        o.y = fmaf((v.y - mu) * is, w.y, g.y);
        o.z = fmaf((v.z - mu) * is, w.z, g.z);
        o.w = fmaf((v.w - mu) * is, w.w, g.w);
        __builtin_nontemporal_store(o, reinterpret_cast<v4f*>(orp + c));
    }
}

// ---------------------------------------------------------------------------
extern "C" void kernel_launch(void* const* d_in, const int* in_sizes, int n_in,
                              void* d_out, int out_size, void* d_ws, size_t ws_size,
                              hipStream_t stream) {
    const float* x    = (const float*)d_in[0];   // [N, 512]
    const float* wgt  = (const float*)d_in[1];   // [1, 512]
    const float* bias = (const float*)d_in[2];   // [1, 512]
    const int*   bidx = (const int*)d_in[3];     // [N], sorted, 0..15
    float*       out  = (float*)d_out;           // [N, 512]

    float* ws   = (float*)d_ws;                  // 80 floats used
    float* s1   = ws;                            // [16]
    float* s2   = ws + 16;                       // [16]
    float* cnt  = ws + 32;                       // [16]
    float* mean = ws + 48;                       // [16]
    float* inv  = ws + 64;                       // [16]

    (void)in_sizes; (void)n_in; (void)out_size; (void)ws_size;

    pln_zero<<<1, 64, 0, stream>>>(ws);
    pln_stats<<<NROWS / (16 * 8), 256, 0, stream>>>(x, bidx, s1, s2, cnt);   // 1024 blocks
    pln_finalize<<<1, NBATCH, 0, stream>>>(ws);
    pln_norm<<<NROWS / 8, 256, 0, stream>>>(x, bidx, wgt, bias, mean, inv, out); // 16384 blocks
}